// SpGQAFlashDecodeAttentionRCCL2_61495341744678
// MI455X (gfx1250) — compile-verified
//
#include <hip/hip_runtime.h>
#include <hip/hip_bf16.h>

typedef __attribute__((ext_vector_type(2))) float v2f;
typedef __attribute__((ext_vector_type(4))) float v4f;
typedef __attribute__((ext_vector_type(8))) float v8f;

#define NUM_RANKS 4
#define BATCH     8
#define HQ        32
#define HKV       8
#define GRP       4            // HQ / HKV
#define DHEAD     128
#define DVHEAD    128
#define PAGE      16
#define MAXP      256
#define NSPLITS   16
#define PPS       (MAXP / NSPLITS)   // pages per split = 16
#define TPS       (PPS * PAGE)       // tokens per split = 256
#define LDSTR     132                // padded LDS row stride (floats), kills bank conflicts
#define PARTW     (DVHEAD + 2)       // per-(head,split) partial: 128 acc + m + l
#define NEGM      (-3.0e38f)
#define QSCALE    0.08838834764831845f   // 1/sqrt(128)

#if __has_builtin(__builtin_amdgcn_s_wait_asynccnt)
#define USE_ASYNC_LDS 1
#else
#define USE_ASYNC_LDS 0
#endif

__device__ __forceinline__ float half_max16(float v) {
  v = fmaxf(v, __shfl_xor(v, 1, 32));
  v = fmaxf(v, __shfl_xor(v, 2, 32));
  v = fmaxf(v, __shfl_xor(v, 4, 32));
  v = fmaxf(v, __shfl_xor(v, 8, 32));
  return v;
}
__device__ __forceinline__ float half_sum16(float v) {
  v += __shfl_xor(v, 1, 32);
  v += __shfl_xor(v, 2, 32);
  v += __shfl_xor(v, 4, 32);
  v += __shfl_xor(v, 8, 32);
  return v;
}

#if USE_ASYNC_LDS
// One async b128 DMA transfer: global -> LDS, no VGPR staging. ASYNCcnt += 1.
// VDST = LDS byte address (low 32 bits of the flat shared pointer),
// VADDR = 64-bit global address, saddr = off (GV mode).
__device__ __forceinline__ void async_ld_b128(unsigned lds_byte_addr, const float* gsrc) {
  asm volatile("global_load_async_to_lds_b128 %0, %1, off"
               :
               : "v"(lds_byte_addr), "v"(gsrc)
               : "memory");
}

// Issue one page of K+V (16 tokens x 128 f32 each) as 32 async b128 DMA loads.
// ASYNCcnt += 32 per call.
__device__ __forceinline__ void issue_page_async(const float* kbase, const float* vbase,
                                                 const float* kdst, const float* vdst,
                                                 int lane) {
  const unsigned koff = (unsigned)(unsigned long long)kdst + (unsigned)lane * 16u;
  const unsigned voff = (unsigned)(unsigned long long)vdst + (unsigned)lane * 16u;
#pragma unroll
  for (int t = 0; t < PAGE; ++t) {
    async_ld_b128(koff + (unsigned)t * (LDSTR * 4u),
                  kbase + (size_t)t * HKV * DHEAD + lane * 4);
    async_ld_b128(voff + (unsigned)t * (LDSTR * 4u),
                  vbase + (size_t)t * HKV * DVHEAD + lane * 4);
  }
}
#endif

// One wave32 per workgroup handles one (batch, kv_head, kv_split).
// QK^T and P*V both run on v_wmma_f32_16x16x4_f32; pages are double-buffered
// through LDS via the async-DMA path so the next page streams during compute.
__global__ __launch_bounds__(32)
void fd_partial(const float* __restrict__ q,
                const float* __restrict__ k_cache,
                const float* __restrict__ v_cache,
                const int*   __restrict__ kv_lens,      // [NUM_RANKS, B]
                const int*   __restrict__ block_table,  // [B, MAXP]
                float*       __restrict__ part) {       // [B*HQ, NSPLITS, PARTW]
  __shared__ float kT[2][PAGE * LDSTR];
  __shared__ float vT[2][PAGE * LDSTR];
  __shared__ float pT[16 * 16];

  const int wg    = blockIdx.x;
  const int split = wg % NSPLITS;
  const int h     = (wg / NSPLITS) % HKV;
  const int b     = wg / (NSPLITS * HKV);
  const int lane  = threadIdx.x;
  const int ln    = lane & 15;
  const int hi    = lane >> 4;

  const int len0 = kv_lens[0 * BATCH + b];
  const int len1 = kv_lens[1 * BATCH + b];
  const int len2 = kv_lens[2 * BATCH + b];
  const int len3 = kv_lens[3 * BATCH + b];
  const int maxlen = max(max(len0, len1), max(len2, len3));
  const int split_start = split * TPS;

  if (split_start >= maxlen) {
    // empty split: zero partial with m = -inf, l = 0
    for (int r = 0; r < GRP; ++r) {
      const size_t base = ((size_t)(b * HQ + h * GRP + r) * NSPLITS + split) * PARTW;
      for (int i = lane; i < PARTW; i += 32)
        part[base + i] = (i == DVHEAD) ? NEGM : 0.f;
    }
    return;
  }

  // zero the P staging tile (rows 4..15 stay zero forever = padded q rows)
  for (int i = lane; i < 256; i += 32) pT[i] = 0.f;

  // Load Q (G=4 real rows, padded to 16) in WMMA A-layout:
  // qa[c] = { Q(m, 4c + 2*hi), Q(m, 4c + 2*hi + 1) }, m = lane&15
  v2f qa[32];
  const float* qrow = q + ((size_t)b * HQ + (size_t)h * GRP + ln) * DHEAD;
  const bool qv = (ln < GRP);
#pragma unroll
  for (int c = 0; c < 32; ++c) {
    v2f z = {0.f, 0.f};
    qa[c] = qv ? *(const v2f*)(qrow + 4 * c + 2 * hi) : z;
  }

  float mrun[GRP], lrun[GRP];
  v8f acc[8];
#pragma unroll
  for (int r = 0; r < GRP; ++r) { mrun[r] = NEGM; lrun[r] = 0.f; }
#pragma unroll
  for (int blk = 0; blk < 8; ++blk) {
    v8f z = {0.f, 0.f, 0.f, 0.f, 0.f, 0.f, 0.f, 0.f};
    acc[blk] = z;
  }

  const int bt_base = b * MAXP + split * PPS;

#if USE_ASYNC_LDS
  {
    const int page0 = block_table[bt_base];
    issue_page_async(k_cache + ((size_t)page0 * PAGE * HKV + h) * DHEAD,
                     v_cache + ((size_t)page0 * PAGE * HKV + h) * DVHEAD,
                     kT[0], vT[0], lane);
  }
  int buf = 0;
#endif

  for (int pi = 0; pi < PPS; ++pi) {
    const int tile_start = split_start + pi * PAGE;
    if (tile_start >= maxlen) break;

#if USE_ASYNC_LDS
    // Kick off next page into the other buffer before computing this one.
    const bool have_next = (pi + 1 < PPS) && (tile_start + PAGE < maxlen);
    if (have_next) {
      const int np = block_table[bt_base + pi + 1];
      issue_page_async(k_cache + ((size_t)np * PAGE * HKV + h) * DHEAD,
                       v_cache + ((size_t)np * PAGE * HKV + h) * DVHEAD,
                       kT[buf ^ 1], vT[buf ^ 1], lane);
      __builtin_amdgcn_s_wait_asynccnt(32);  // current page resident, next in flight
    } else {
      __builtin_amdgcn_s_wait_asynccnt(0);
    }
    __syncthreads();
    const float* kCur = kT[buf];
    const float* vCur = vT[buf];
#else
    const int page = block_table[bt_base + pi];
    const float* kbase = k_cache + ((size_t)page * PAGE * HKV + h) * DHEAD;
    const float* vbase = v_cache + ((size_t)page * PAGE * HKV + h) * DVHEAD;
#pragma unroll
    for (int t = 0; t < PAGE; ++t) {
      *(v4f*)&kT[0][t * LDSTR + lane * 4] = *(const v4f*)(kbase + (size_t)t * HKV * DHEAD + lane * 4);
      *(v4f*)&vT[0][t * LDSTR + lane * 4] = *(const v4f*)(vbase + (size_t)t * HKV * DVHEAD + lane * 4);
    }
    if (pi + 1 < PPS && tile_start + PAGE < maxlen) {
      const int np = block_table[bt_base + pi + 1];
      __builtin_prefetch(k_cache + ((size_t)np * PAGE * HKV + h) * DHEAD + lane * 4, 0, 0);
      __builtin_prefetch(v_cache + ((size_t)np * PAGE * HKV + h) * DVHEAD + lane * 4, 0, 0);
    }
    __syncthreads();
    const float* kCur = kT[0];
    const float* vCur = vT[0];
#endif

    // S = Q * K^T : M=16 (q rows), N=16 (tokens), K=128 -> 32 x wmma 16x16x4 f32
    v8f s = {0.f, 0.f, 0.f, 0.f, 0.f, 0.f, 0.f, 0.f};
    const float* kRow = &kCur[ln * LDSTR + 2 * hi];  // B-layout: row K, col = token lane
#pragma unroll
    for (int c = 0; c < 32; ++c) {
      v2f kb = *(const v2f*)(kRow + 4 * c);
      s = __builtin_amdgcn_wmma_f32_16x16x4_f32(false, qa[c], false, kb,
                                                (short)0, s, false, false);
    }

    // Per-token rank multiplicity folded into a log-bias:
    // combined softmax weight = cnt * exp(score) = exp(score + log(cnt))
    const int tok = tile_start + ln;
    const int cnt = (tok < len0) + (tok < len1) + (tok < len2) + (tok < len3);
    const float validf = (cnt > 0) ? 1.f : 0.f;
    const float bias = __logf(fmaxf((float)cnt, 1.f));

#pragma unroll
    for (int r = 0; r < GRP; ++r) {
      float lg = (cnt > 0) ? (s[r] * QSCALE + bias) : NEGM;
      float rm = half_max16(lg);
      float mnew = fmaxf(mrun[r], rm);
      float alpha = __expf(mrun[r] - mnew);
      float p = validf * __expf(lg - mnew);
      float rs = half_sum16(p);
      lrun[r] = lrun[r] * alpha + rs;
      mrun[r] = mnew;
#pragma unroll
      for (int blk = 0; blk < 8; ++blk) acc[blk][r] *= alpha;
      if (lane < 16) pT[r * 16 + lane] = p;  // rows 4..15 remain zero
    }
    __syncthreads();

    // O += P * V : 8 dv-blocks of 16, K=16 tokens -> 32 x wmma 16x16x4 f32
#pragma unroll
    for (int kc = 0; kc < 4; ++kc) {
      const int t0 = kc * 4 + 2 * hi;
      v2f pa = *(const v2f*)&pT[ln * 16 + t0];           // A-layout P chunk
      const float* v0 = &vCur[(size_t)t0 * LDSTR + ln];  // B-layout V chunk
#pragma unroll
      for (int blk = 0; blk < 8; ++blk) {
        v2f vb;
        vb.x = v0[blk * 16];
        vb.y = v0[blk * 16 + LDSTR];
        acc[blk] = __builtin_amdgcn_wmma_f32_16x16x4_f32(false, pa, false, vb,
                                                         (short)0, acc[blk], false, false);
      }
    }
    __syncthreads();

#if USE_ASYNC_LDS
    buf ^= 1;
#endif
  }

  // Emit per-split partials (unnormalized acc, running max m, running sum l).
#pragma unroll
  for (int r = 0; r < GRP; ++r) {
    const size_t base = ((size_t)(b * HQ + h * GRP + r) * NSPLITS + split) * PARTW;
    if (lane < 16) {
#pragma unroll
      for (int blk = 0; blk < 8; ++blk)
        part[base + blk * 16 + lane] = acc[blk][r];
    }
    if (lane == 0) {
      part[base + DVHEAD]     = mrun[r];
      part[base + DVHEAD + 1] = lrun[r];
    }
  }
}

// LSE combine across the NSPLITS partials. One block per (b, q_head).
__global__ __launch_bounds__(DVHEAD)
void fd_combine(const float* __restrict__ part, float* __restrict__ out) {
  const int row = blockIdx.x;     // b*HQ + hq
  const int d   = threadIdx.x;

  float M = NEGM;
#pragma unroll
  for (int s = 0; s < NSPLITS; ++s)
    M = fmaxf(M, part[((size_t)row * NSPLITS + s) * PARTW + DVHEAD]);

  float L = 0.f, o = 0.f;
#pragma unroll
  for (int s = 0; s < NSPLITS; ++s) {
    const size_t base = ((size_t)row * NSPLITS + s) * PARTW;
    const float w = __expf(part[base + DVHEAD] - M);   // 0 for empty splits
    L += part[base + DVHEAD + 1] * w;
    o += part[base + d] * w;
  }
  out[(size_t)row * DVHEAD + d] = o / L;
}

extern "C" void kernel_launch(void* const* d_in, const int* in_sizes, int n_in,
                              void* d_out, int out_size, void* d_ws, size_t ws_size,
                              hipStream_t stream) {
  (void)in_sizes; (void)n_in; (void)out_size; (void)ws_size;
  const float* q           = (const float*)d_in[0];
  const float* k_cache     = (const float*)d_in[1];
  const float* v_cache     = (const float*)d_in[2];
  const int*   kv_lens     = (const int*)d_in[3];
  const int*   block_table = (const int*)d_in[4];
  float* partials          = (float*)d_ws;   // B*HQ*NSPLITS*PARTW floats (~2.1 MB)

  fd_partial<<<dim3(BATCH * HKV * NSPLITS), dim3(32), 0, stream>>>(
      q, k_cache, v_cache, kv_lens, block_table, partials);
  fd_combine<<<dim3(BATCH * HQ), dim3(DVHEAD), 0, stream>>>(
      partials, (float*)d_out);
}